// PositionalEmbedding_9491877724363
// MI455X (gfx1250) — compile-verified
//
#include <hip/hip_runtime.h>
#include <hip/hip_bf16.h>
#include <stdint.h>

// PositionalEmbedding: out[l, :] = emb_table[token_ids[l], :] + sinusoidal_PE(l, :)
//   SEQ = 131072, DEPTH = 512, table = 50257 x 512 f32 (~103 MiB, fits MI455X 192MB L2)
//
// Bandwidth-bound: ~256 MiB NT stores + ~103 MiB compulsory table reads -> ~16-22us
// floor at 23.3 TB/s. CDNA5 paths used: global_load_async_to_lds_b128 (ASYNCcnt)
// double-buffered row staging overlapped with v_sin/v_cos PE compute, s_wait_asynccnt,
// and non-temporal global_store_b128 so the output stream does not evict the table
// from L2.

#define SEQ_LEN 131072
#define DEPTH   512
#define TPB     128   // 4 wave32s; each thread owns one float4 (4 columns)

typedef __attribute__((ext_vector_type(4))) float v4f;

__global__ void __launch_bounds__(TPB) pos_emb_kernel(
    const int*   __restrict__ token_ids,
    const float* __restrict__ emb_table,
    float*       __restrict__ out)
{
    // Double buffer: 2 x 2KB. Each lane stages and later reads ONLY its own
    // 16 bytes, so the only synchronization needed is the async counter wait
    // (no workgroup barrier).
    __shared__ v4f buf[2][TPB];

    const int tid = threadIdx.x;

    // This thread's two (sin, cos) column pairs: j0 = 2*tid, j1 = 2*tid + 1.
    // inv_freq(j) = 10000^(-2j/512) = exp2(-j * log2(10000)/256)
    const int   j0 = 2 * tid;
    const int   j1 = 2 * tid + 1;
    const float K  = 13.287712379549449f / 256.0f;   // log2(10000)/256
    const float invf0 = __builtin_amdgcn_exp2f(-K * (float)j0);  // v_exp_f32
    const float invf1 = __builtin_amdgcn_exp2f(-K * (float)j1);  // v_exp_f32
    // Reference only fills 255 pairs: columns 510/511 stay zero -> mask pair 255.
    const float m1 = (j1 < 255) ? 1.0f : 0.0f;

    const unsigned stride = gridDim.x;
    unsigned row = blockIdx.x;
    int cur = 0;

    // Prologue: stage first row into buf[0] via the async copy engine.
    if (row < SEQ_LEN) {
        const int    tok = token_ids[row];          // block-uniform -> s_load
        const float* src = emb_table + (size_t)tok * DEPTH + 4 * tid;
        unsigned lds = (unsigned)(uintptr_t)&buf[0][tid];
        asm volatile("global_load_async_to_lds_b128 %0, %1, off"
                     :: "v"(lds), "v"(src) : "memory");
    }

    for (; row < SEQ_LEN; row += stride) {
        const unsigned nrow = row + stride;
        const bool has_next = (nrow < SEQ_LEN);

        // Stage next row into the other buffer while we work on this one.
        if (has_next) {
            const int    ntok = token_ids[nrow];
            const float* nsrc = emb_table + (size_t)ntok * DEPTH + 4 * tid;
            unsigned nlds = (unsigned)(uintptr_t)&buf[cur ^ 1][tid];
            asm volatile("global_load_async_to_lds_b128 %0, %1, off"
                         :: "v"(nlds), "v"(nsrc) : "memory");
        }

        // Positional encoding for this row: hidden under the in-flight DMA.
        const float p  = (float)row;
        const float a0 = p * invf0;
        const float a1 = p * invf1;
        const float s0 = __sinf(a0);
        const float c0 = __cosf(a0);
        const float s1 = __sinf(a1) * m1;
        const float c1 = __cosf(a1) * m1;

        // Wait for THIS row's async load (loads complete in order: <=1 leaves
        // only the just-issued next-row load outstanding).
        if (has_next) {
            asm volatile("s_wait_asynccnt 0x1" ::: "memory");
        } else {
            asm volatile("s_wait_asynccnt 0x0" ::: "memory");
        }

        v4f e = buf[cur][tid];                       // ds_load_b128
        e.x += s0;
        e.y += c0;
        e.z += s1;
        e.w += c1;

        // Non-temporal store: keep the 103 MiB table resident in L2, stream
        // the 256 MiB output past it.
        __builtin_nontemporal_store(
            e, (v4f*)(out + (size_t)row * DEPTH + 4 * tid));

        cur ^= 1;
    }
}

extern "C" void kernel_launch(void* const* d_in, const int* in_sizes, int n_in,
                              void* d_out, int out_size, void* d_ws, size_t ws_size,
                              hipStream_t stream) {
    const int*   token_ids = (const int*)d_in[0];    // [131072] int32
    const float* emb_table = (const float*)d_in[1];  // [50257, 512] f32
    float*       out       = (float*)d_out;          // [131072, 512] f32

    // 8192 blocks x 16 rows each: enough blocks to saturate every WGP while
    // amortizing the per-thread exp2 inverse-frequency setup over 16 rows.
    dim3 grid(8192), block(TPB);
    hipLaunchKernelGGL(pos_emb_kernel, grid, block, 0, stream,
                       token_ids, emb_table, out);
}